// CRFLayer_65120294142164
// MI455X (gfx1250) — compile-verified
//
#include <hip/hip_runtime.h>
#include <math.h>

typedef __attribute__((ext_vector_type(2))) float v2f;
typedef __attribute__((ext_vector_type(8))) float v8f;

#define HW    512
#define NPIX  (HW * HW)
#define NC    21

// ---------------------------------------------------------------- utilities
__global__ void zero_f32(float* __restrict__ p, long long n) {
    long long t = (long long)blockIdx.x * blockDim.x + threadIdx.x;
    if (t < n) p[t] = 0.0f;
}

// ------------------------------------------------------------- norm stage (C=1)
__global__ void splat_ones(const float* __restrict__ ws, const int* __restrict__ os,
                           float* __restrict__ val1, int d1) {
    int n = blockIdx.x * blockDim.x + threadIdx.x;
    if (n >= NPIX) return;
    for (int j = 0; j < d1; ++j)
        atomicAdd(&val1[os[n * d1 + j]], ws[n * d1 + j]);
}

__global__ void blur_c1(const float* __restrict__ in, float* __restrict__ out,
                        const int* __restrict__ nbr, int M, int j) {
    int m = blockIdx.x * blockDim.x + threadIdx.x;
    if (m >= M) return;
    int n1 = nbr[((long long)j * M + m) * 2 + 0];
    int n2 = nbr[((long long)j * M + m) * 2 + 1];
    out[m + 1] = in[m + 1] + 0.5f * (in[n1] + in[n2]);   // row 0 = zero pad
}

__global__ void slice_norm(const float* __restrict__ ws, const int* __restrict__ os,
                           const float* __restrict__ val1, float* __restrict__ norm,
                           int d1, float alpha) {
    int n = blockIdx.x * blockDim.x + threadIdx.x;
    if (n >= NPIX) return;
    float s = 0.0f;
    for (int j = 0; j < d1; ++j)
        s += ws[n * d1 + j] * val1[os[n * d1 + j]];
    s *= alpha;
    norm[n] = 1.0f / (sqrtf(s) + 1e-20f);
}

// ------------------------------------------------------------- splat (C=21)
__global__ void splat_q(const float* __restrict__ Q, const float* __restrict__ norm,
                        const float* __restrict__ ws, const int* __restrict__ os,
                        float* __restrict__ val, int d1) {
    int n = blockIdx.x * blockDim.x + threadIdx.x;
    if (n >= NPIX) return;
    // gfx1250 prefetch of upcoming Q rows (speculative; dropped if OOB translate)
    __builtin_prefetch(&Q[(size_t)(n + 4 * 256) * NC], 0, 1);
    float nv = norm[n];
    float q[NC];
#pragma unroll
    for (int c = 0; c < NC; ++c) q[c] = Q[(size_t)n * NC + c] * nv;
    for (int j = 0; j < d1; ++j) {
        int   m = os[n * d1 + j];
        float w = ws[n * d1 + j];
        float* row = val + (size_t)m * NC;
#pragma unroll
        for (int c = 0; c < NC; ++c) atomicAdd(&row[c], w * q[c]);
    }
}

// ------------------------------------------------------------- blur (C=21)
__global__ void blur_cN(const float* __restrict__ in, float* __restrict__ out,
                        const int* __restrict__ nbr, int M, int j) {
    long long t = (long long)blockIdx.x * blockDim.x + threadIdx.x;
    long long total = (long long)M * NC;
    if (t >= total) return;
    int m = (int)(t / NC);
    int c = (int)(t - (long long)m * NC);
    int n1 = nbr[((long long)j * M + m) * 2 + 0];
    int n2 = nbr[((long long)j * M + m) * 2 + 1];
    out[(size_t)(m + 1) * NC + c] =
        in[(size_t)(m + 1) * NC + c] +
        0.5f * (in[(size_t)n1 * NC + c] + in[(size_t)n2 * NC + c]);
}

// ------------------------------------------------------------- slice via WMMA
// out16[p,c] = sum_j ws[p,j] * val[os[p,j], c]  for 16 pixels/wave.
// = sum_j sum_kk A(16x4) x B(4x16):
//   A[m,k'] = ws[m,j] * (m == 4*kk+k')   (diagonal chunk)
//   B[k',n] = val[os[4*kk+k', j], ct*16+n]
// Accumulated with V_WMMA_F32_16X16X4_F32 into v8f; EXEC stays all-ones
// (no divergent branches before the WMMAs; masking via selects/mults).
__global__ void slice_wmma(const float* __restrict__ val, const float* __restrict__ ws,
                           const int* __restrict__ os, const float* __restrict__ norm,
                           float* __restrict__ pw, int d1, float scale, int accum) {
    int lane = threadIdx.x & 31;
    int wave = threadIdx.x >> 5;
    int base = (blockIdx.x * (blockDim.x >> 5) + wave) * 16;   // 16 pixels per wave
    int mrow = lane & 15;
    int hi   = lane >> 4;

    v8f acc0 = {};   // channels 0..15
    v8f acc1 = {};   // channels 16..20 (cols 21..31 masked to 0)

    int c0  = mrow;
    int c1  = 16 + mrow;
    int c1c = (c1 > NC - 1) ? (NC - 1) : c1;
    float c1ok = (c1 < NC) ? 1.0f : 0.0f;

    for (int j = 0; j < d1; ++j) {
        float wmj = ws[(base + mrow) * d1 + j];
#pragma unroll
        for (int kk = 0; kk < 4; ++kk) {
            int k0 = 4 * kk + (hi ? 2 : 0);
            int k1 = 4 * kk + (hi ? 3 : 1);
            v2f a;
            a.x = (mrow == k0) ? wmj : 0.0f;
            a.y = (mrow == k1) ? wmj : 0.0f;
            int r0 = os[(base + k0) * d1 + j];
            int r1 = os[(base + k1) * d1 + j];
            const float* row0 = val + (size_t)r0 * NC;
            const float* row1 = val + (size_t)r1 * NC;
            v2f b0, b1;
            b0.x = row0[c0];        b0.y = row1[c0];
            b1.x = row0[c1c] * c1ok; b1.y = row1[c1c] * c1ok;
            acc0 = __builtin_amdgcn_wmma_f32_16x16x4_f32(false, a, false, b0,
                                                         (short)0, acc0, false, false);
            acc1 = __builtin_amdgcn_wmma_f32_16x16x4_f32(false, a, false, b1,
                                                         (short)0, acc1, false, false);
        }
    }

    // D layout: VGPR r -> (M = r + 8*hi, N = mrow). Scale by alpha*compat*norm[p].
#pragma unroll
    for (int r = 0; r < 8; ++r) {
        int   p = base + r + (hi ? 8 : 0);
        float s = scale * norm[p];
        size_t i0 = (size_t)p * NC + c0;
        float v0 = acc0[r] * s;
        pw[i0] = accum ? (pw[i0] + v0) : v0;
        if (c1 < NC) {
            size_t i1 = (size_t)p * NC + c1;
            float v1 = acc1[r] * s;
            pw[i1] = accum ? (pw[i1] + v1) : v1;
        }
    }
}

// ------------------------------------------------------------- softmax
__global__ void softmax_k(const float* __restrict__ u, const float* __restrict__ pw,
                          float* __restrict__ Q, int usePw) {
    int n = blockIdx.x * blockDim.x + threadIdx.x;
    if (n >= NPIX) return;
    float l[NC];
    float mx = -3.0e38f;
#pragma unroll
    for (int c = 0; c < NC; ++c) {
        float v = -u[(size_t)n * NC + c];
        if (usePw) v += pw[(size_t)n * NC + c];
        l[c] = v;
        mx = fmaxf(mx, v);
    }
    float s = 0.0f;
#pragma unroll
    for (int c = 0; c < NC; ++c) {
        float e = expf(l[c] - mx);
        l[c] = e;
        s += e;
    }
    float inv = 1.0f / s;
#pragma unroll
    for (int c = 0; c < NC; ++c) Q[(size_t)n * NC + c] = l[c] * inv;
}

// ------------------------------------------------------------- host driver
extern "C" void kernel_launch(void* const* d_in, const int* in_sizes, int n_in,
                              void* d_out, int out_size, void* d_ws, size_t ws_size,
                              hipStream_t stream) {
    const float* unary  = (const float*)d_in[0];
    const float* ws_bi  = (const float*)d_in[1];
    const float* ws_sp  = (const float*)d_in[2];
    const int*   os_bi  = (const int*)d_in[3];
    const int*   os_sp  = (const int*)d_in[4];
    const int*   nbr_bi = (const int*)d_in[5];
    const int*   nbr_sp = (const int*)d_in[6];
    float*       Q      = (float*)d_out;

    const int d1_bi = in_sizes[1] / NPIX;          // d+1 = 6
    const int d1_sp = in_sizes[2] / NPIX;          // d+1 = 3
    const int M_bi  = in_sizes[5] / (2 * d1_bi);
    const int M_sp  = in_sizes[6] / (2 * d1_sp);
    const int Mmax  = (M_bi > M_sp) ? M_bi : M_sp;

    const float alpha_bi = 1.0f / (1.0f + exp2f(-(float)(d1_bi - 1)));
    const float alpha_sp = 1.0f / (1.0f + exp2f(-(float)(d1_sp - 1)));

    // scratch layout
    char*  wsb = (char*)d_ws;
    size_t off = 0;
    auto alloc = [&](size_t bytes) -> void* {
        void* p = wsb + off;
        off += (bytes + 255) & ~(size_t)255;
        return p;
    };
    float* pw      = (float*)alloc((size_t)NPIX * NC * sizeof(float));
    float* norm_bi = (float*)alloc((size_t)NPIX * sizeof(float));
    float* norm_sp = (float*)alloc((size_t)NPIX * sizeof(float));
    float* valA    = (float*)alloc((size_t)(Mmax + 1) * NC * sizeof(float));
    float* valB    = (float*)alloc((size_t)(Mmax + 1) * NC * sizeof(float));
    (void)ws_size; (void)n_in; (void)out_size;

    const int B = 256;
    dim3 blk(B);
    auto g1 = [&](long long n) { return dim3((unsigned)((n + B - 1) / B)); };

    // ---------- normalization factors (C = 1) ----------
    // bilateral
    zero_f32<<<g1(M_bi + 1), blk, 0, stream>>>(valA, M_bi + 1);
    zero_f32<<<g1(M_bi + 1), blk, 0, stream>>>(valB, M_bi + 1);
    splat_ones<<<g1(NPIX), blk, 0, stream>>>(ws_bi, os_bi, valA, d1_bi);
    {
        float* s = valA; float* d = valB;
        for (int j = 0; j < d1_bi; ++j) {
            blur_c1<<<g1(M_bi), blk, 0, stream>>>(s, d, nbr_bi, M_bi, j);
            float* t = s; s = d; d = t;
        }
        slice_norm<<<g1(NPIX), blk, 0, stream>>>(ws_bi, os_bi, s, norm_bi, d1_bi, alpha_bi);
    }
    // spatial
    zero_f32<<<g1(M_sp + 1), blk, 0, stream>>>(valA, M_sp + 1);
    zero_f32<<<g1(M_sp + 1), blk, 0, stream>>>(valB, M_sp + 1);
    splat_ones<<<g1(NPIX), blk, 0, stream>>>(ws_sp, os_sp, valA, d1_sp);
    {
        float* s = valA; float* d = valB;
        for (int j = 0; j < d1_sp; ++j) {
            blur_c1<<<g1(M_sp), blk, 0, stream>>>(s, d, nbr_sp, M_sp, j);
            float* t = s; s = d; d = t;
        }
        slice_norm<<<g1(NPIX), blk, 0, stream>>>(ws_sp, os_sp, s, norm_sp, d1_sp, alpha_sp);
    }

    // ---------- Q0 = softmax(-u) ----------
    softmax_k<<<g1(NPIX), blk, 0, stream>>>(unary, pw, Q, 0);

    // ---------- mean-field iterations ----------
    const int sliceBlocks = NPIX / (16 * (B / 32));   // 16 pixels per wave, 8 waves/block
    for (int it = 0; it < 5; ++it) {
        // bilateral lattice: pairwise = 10 * bi
        zero_f32<<<g1((long long)(M_bi + 1) * NC), blk, 0, stream>>>(valA, (long long)(M_bi + 1) * NC);
        zero_f32<<<g1(NC), blk, 0, stream>>>(valB, NC);           // zero pad row of ping buffer
        splat_q<<<g1(NPIX), blk, 0, stream>>>(Q, norm_bi, ws_bi, os_bi, valA, d1_bi);
        {
            float* s = valA; float* d = valB;
            for (int j = 0; j < d1_bi; ++j) {
                blur_cN<<<g1((long long)M_bi * NC), blk, 0, stream>>>(s, d, nbr_bi, M_bi, j);
                float* t = s; s = d; d = t;
            }
            slice_wmma<<<dim3(sliceBlocks), blk, 0, stream>>>(s, ws_bi, os_bi, norm_bi, pw,
                                                              d1_bi, 10.0f * alpha_bi, 0);
        }
        // spatial lattice: pairwise += 3 * sp
        zero_f32<<<g1((long long)(M_sp + 1) * NC), blk, 0, stream>>>(valA, (long long)(M_sp + 1) * NC);
        zero_f32<<<g1(NC), blk, 0, stream>>>(valB, NC);
        splat_q<<<g1(NPIX), blk, 0, stream>>>(Q, norm_sp, ws_sp, os_sp, valA, d1_sp);
        {
            float* s = valA; float* d = valB;
            for (int j = 0; j < d1_sp; ++j) {
                blur_cN<<<g1((long long)M_sp * NC), blk, 0, stream>>>(s, d, nbr_sp, M_sp, j);
                float* t = s; s = d; d = t;
            }
            slice_wmma<<<dim3(sliceBlocks), blk, 0, stream>>>(s, ws_sp, os_sp, norm_sp, pw,
                                                              d1_sp, 3.0f * alpha_sp, 1);
        }
        // Q = softmax(-u + pairwise)
        softmax_k<<<g1(NPIX), blk, 0, stream>>>(unary, pw, Q, 1);
    }
}